// BeamSearchDecoder_68307159876045
// MI455X (gfx1250) — compile-verified
//
#include <hip/hip_runtime.h>
#include <stdint.h>

#define BB 64
#define TT 1024
#define CC 1024
#define KK 8

// ---- order-preserving float <-> uint32 (ascending float order == ascending uint order) ----
__device__ __forceinline__ uint32_t ordf(float f) {
  uint32_t s = __float_as_uint(f);
  uint32_t mask = (uint32_t)((int32_t)s >> 31) | 0x80000000u;
  return s ^ mask;
}
__device__ __forceinline__ float iordf(uint32_t u) {
  uint32_t s = (u & 0x80000000u) ? (u ^ 0x80000000u) : ~u;
  return __uint_as_float(s);
}

// ---- async global->LDS stage of one 4KB row (1024 floats): 8 x b128 per lane ----
// mapping: class c = j*128 + lane*4 + e  (contiguous 512B per instruction, fully coalesced)
__device__ __forceinline__ void issue_row_async(const float* rowp, int lane, float* dstbuf) {
  unsigned long long sbase = (unsigned long long)(uintptr_t)rowp;
#pragma unroll
  for (int j = 0; j < 8; ++j) {
    unsigned voff  = (unsigned)(j * 512 + lane * 16);                       // global byte offset
    unsigned laddr = (unsigned)(uintptr_t)(dstbuf + j * 128 + lane * 4);    // LDS byte offset (addr[31:0])
    asm volatile("global_load_async_to_lds_b128 %0, %1, %2"
                 :: "v"(laddr), "v"(voff), "s"(sbase)
                 : "memory");
  }
}

// ================= Kernel 1: beam search, one wave32 per batch row =================
__global__ __launch_bounds__(32) void beam_kernel(const float* __restrict__ probs,
                                                  int* __restrict__ seq) {
  __shared__ __align__(16) float buf[2][CC];       // 8 KB double buffer (async target)
  __shared__ unsigned short bp[TT * KK];           // 16 KB backpointers: (parent<<10)|class

  const int lane = (int)threadIdx.x;
  const int b    = (int)blockIdx.x;
  const float* rowp0 = probs + (size_t)b * TT * CC;
  const float INF  = __uint_as_float(0x7F800000u);
  const float NINF = __uint_as_float(0xFF800000u);

  // scores distributed: lane q, q+8, q+16, q+24 hold beam q's score (replicated x4)
  float myscore = ((lane & 7) == 0) ? 0.0f : INF;

  issue_row_async(rowp0, lane, &buf[0][0]);

  for (int t = 0; t < TT; ++t) {
    const int cur = t & 1;
    const int nt  = (t + 1 < TT) ? (t + 1) : (TT - 1);   // clamp: harmless redundant prefetch
    issue_row_async(rowp0 + (size_t)nt * CC, lane, &buf[(t + 1) & 1][0]);
    asm volatile("s_wait_asynccnt 0x8" ::: "memory");    // previous row's 8 copies landed

    // ---- phase 1: top-8 largest row values (tie -> lowest class), exact ----
    float rv_mine = 0.0f;   // distributed by (lane&7): r-th largest row value
    int   wc_mine = 0;      // its class
    const float4* fb = reinterpret_cast<const float4*>(&buf[cur][0]);
#pragma unroll 1
    for (int r = 0; r < KK; ++r) {
      uint32_t m = 0u; int a = 0;
#pragma unroll
      for (int j = 0; j < 8; ++j) {
        float4 q = fb[j * 32 + lane];
        uint32_t u0 = ordf(q.x), u1 = ordf(q.y), u2 = ordf(q.z), u3 = ordf(q.w);
        if (u0 > m) { m = u0; a = j * 4 + 0; }   // strict '>' keeps lowest index,
        if (u1 > m) { m = u1; a = j * 4 + 1; }   // and class is monotonic in index
        if (u2 > m) { m = u2; a = j * 4 + 2; }
        if (u3 > m) { m = u3; a = j * 4 + 3; }
      }
      int c = ((a >> 2) << 7) | (lane << 2) | (a & 3);
      unsigned long long key = ((unsigned long long)m << 10) | (unsigned)(1023 - c);
#pragma unroll
      for (int off = 16; off > 0; off >>= 1) {
        unsigned long long o = __shfl_xor(key, off, 32);
        if (o > key) key = o;                    // max value, tie -> lowest class
      }
      uint32_t wu = (uint32_t)(key >> 10);
      int wc = 1023 - (int)(key & 1023ull);
      if ((lane & 7) == r) { rv_mine = iordf(wu); wc_mine = wc; }
      if (lane == 0) buf[cur][wc] = NINF;        // mask winner (all finite data > -inf)
    }

    // ---- phase 2: 64 candidates (8 parents x 8 row values), extract 8 smallest ----
    // lane handles candidates (p0, lane&7) and (p0+4, lane&7); tie-break == flat index p*C+c
    int   p0 = lane >> 3;
    float s0 = __shfl(myscore, p0, 32);          // beam p0 score lives on lane p0
    float s1 = __shfl(myscore, p0 + 4, 32);
    unsigned long long k0 = ((unsigned long long)ordf(s0 - rv_mine) << 13)
                          | ((unsigned)p0 << 10) | (unsigned)wc_mine;
    unsigned long long k1 = ((unsigned long long)ordf(s1 - rv_mine) << 13)
                          | ((unsigned)(p0 + 4) << 10) | (unsigned)wc_mine;
    float newscore = myscore;
#pragma unroll 1
    for (int r = 0; r < KK; ++r) {
      unsigned long long w = (k0 < k1) ? k0 : k1;
#pragma unroll
      for (int off = 16; off > 0; off >>= 1) {
        unsigned long long o = __shfl_xor(w, off, 32);
        if (o < w) w = o;                        // min value, tie -> lowest (p,c)
      }
      if (k0 == w) k0 = ~0ull;                   // keys unique: winner masks itself
      if (k1 == w) k1 = ~0ull;
      if ((lane & 7) == r) newscore = iordf((uint32_t)(w >> 13));  // ascending order
      if (lane == r) bp[t * KK + r] = (unsigned short)(w & 0x1FFFull);
    }
    myscore = newscore;
  }

  asm volatile("s_wait_asynccnt 0x0" ::: "memory");  // drain the final dummy prefetch

  // ---- traceback of beam k-1 (the reference returns seqs[k-1]) ----
  if (lane == 0) {
    int beam = KK - 1;
    for (int t = TT - 1; t >= 0; --t) {
      unsigned pk = bp[t * KK + beam];
      seq[(size_t)b * TT + t] = (int)(pk & 1023u);
      beam = (int)(pk >> 10);
    }
  }
}

// ================= Kernel 2: batch-wide consecutive-column dedup mask =================
__global__ void colkeep_kernel(const int* __restrict__ seq, int* __restrict__ colkeep) {
  int t = (int)(blockIdx.x * blockDim.x + threadIdx.x);
  if (t >= TT) return;
  int d = (t == 0) ? 1 : 0;
  if (t > 0) {
    for (int b = 0; b < BB; ++b)
      d |= (seq[(size_t)b * TT + t] != seq[(size_t)b * TT + t - 1]);
  }
  colkeep[t] = d;
}

// ================= Kernel 3: per-row blank removal + stable left-compaction =================
__global__ __launch_bounds__(32) void compact_kernel(const int* __restrict__ seq,
                                                     const int* __restrict__ colkeep,
                                                     float* __restrict__ out) {
  const int b = (int)blockIdx.x, lane = (int)threadIdx.x;
  const int* row = seq + (size_t)b * TT;
  float* dec  = out + (size_t)b * TT;
  float* lens = out + (size_t)BB * TT;

  const int base = lane * 32;
  int cnt = 0;
  for (int i = 0; i < 32; ++i) {
    int t = base + i;
    cnt += (colkeep[t] && row[t] != 0) ? 1 : 0;
  }
  // wave32 inclusive scan -> exclusive offset
  int x = cnt;
#pragma unroll
  for (int d = 1; d < 32; d <<= 1) {
    int y = __shfl_up(x, d, 32);
    if (lane >= d) x += y;
  }
  int pos   = x - cnt;
  int total = __shfl(x, 31, 32);
  for (int i = 0; i < 32; ++i) {
    int t = base + i;
    int v = row[t];
    if (colkeep[t] && v != 0) dec[pos++] = (float)v;
  }
  for (int i = total + lane; i < TT; i += 32) dec[i] = -1.0f;   // pad tail
  if (lane == 0) lens[b] = (float)total;
}

extern "C" void kernel_launch(void* const* d_in, const int* in_sizes, int n_in,
                              void* d_out, int out_size, void* d_ws, size_t ws_size,
                              hipStream_t stream) {
  (void)in_sizes; (void)n_in; (void)out_size; (void)ws_size;
  const float* probs = (const float*)d_in[0];
  float* out = (float*)d_out;
  int* seq     = (int*)d_ws;                 // B*T ints   (256 KB)
  int* colkeep = seq + (size_t)BB * TT;      // T ints     (4 KB)

  hipLaunchKernelGGL(beam_kernel,   dim3(BB),      dim3(32),  0, stream, probs, seq);
  hipLaunchKernelGGL(colkeep_kernel,dim3(TT/256),  dim3(256), 0, stream, seq, colkeep);
  hipLaunchKernelGGL(compact_kernel,dim3(BB),      dim3(32),  0, stream, seq, colkeep, out);
}